// GraphConvolution_4587025072811
// MI455X (gfx1250) — compile-verified
//
#include <hip/hip_runtime.h>

typedef __attribute__((ext_vector_type(2))) float v2f;
typedef __attribute__((ext_vector_type(8))) float v8f;

#define GCN_N_NODES 50000
#define GCN_N_EDGES 800000
#define GCN_IN_F    128
#define GCN_OUT_F   64
#define GCN_M_TILES (GCN_N_NODES / 16)   // 3125, exact

// ---------------------------------------------------------------------------
// Kernel 1: support[N,64] = x[N,128] @ W[128,64] via V_WMMA_F32_16X16X4_F32.
// One wave computes a 16(M) x 64(N) strip: 4 N-tiles of 16x16, K unrolled in
// steps of 4 (32 steps x 4 WMMA = 128 wmma per wave). A-fragment is shared by
// all 4 N-tile WMMAs each step.
//
// Lane decomposition: m = lane & 15, half = lane >> 4.
//   A  frag  (16x4 ):  a[g]   = x[m0+m][k0 + 2*half + g]      (contiguous b64)
//   B  frag  (4x16 ):  b[g]   = W[k0 + 2*half + g][n0 + m]
//   C/D frag (16x16):  acc[g] = D[m0 + g + 8*half][n0 + m]
// ---------------------------------------------------------------------------
__global__ __launch_bounds__(256) void gcn_gemm_wmma(
    const float* __restrict__ x,
    const float* __restrict__ W,
    float* __restrict__ support)
{
    const int lane      = threadIdx.x & 31;
    const int waveInBlk = threadIdx.x >> 5;
    const int strip     = blockIdx.x * 8 + waveInBlk;   // 16-row M strip
    if (strip >= GCN_M_TILES) return;                   // wave-uniform: EXEC stays all-1s

    const int m    = lane & 15;
    const int half = lane >> 4;
    const int row  = strip * 16 + m;

    v8f acc0 = {}; v8f acc1 = {}; v8f acc2 = {}; v8f acc3 = {};

    const float* xr = x + (size_t)row * GCN_IN_F + 2 * half;

    #pragma unroll
    for (int k0 = 0; k0 < GCN_IN_F; k0 += 4) {
        v2f a = *(const v2f*)(xr + k0);                 // A[m][k0+2h .. k0+2h+1]
        const int krow = k0 + 2 * half;
        const float* wp = W + (size_t)krow * GCN_OUT_F + m;
        v2f b0 = { wp[0],  wp[GCN_OUT_F + 0]  };        // N-tile 0  (cols  0..15)
        v2f b1 = { wp[16], wp[GCN_OUT_F + 16] };        // N-tile 1  (cols 16..31)
        v2f b2 = { wp[32], wp[GCN_OUT_F + 32] };        // N-tile 2  (cols 32..47)
        v2f b3 = { wp[48], wp[GCN_OUT_F + 48] };        // N-tile 3  (cols 48..63)
        acc0 = __builtin_amdgcn_wmma_f32_16x16x4_f32(false, a, false, b0, (short)0, acc0, false, false);
        acc1 = __builtin_amdgcn_wmma_f32_16x16x4_f32(false, a, false, b1, (short)0, acc1, false, false);
        acc2 = __builtin_amdgcn_wmma_f32_16x16x4_f32(false, a, false, b2, (short)0, acc2, false, false);
        acc3 = __builtin_amdgcn_wmma_f32_16x16x4_f32(false, a, false, b3, (short)0, acc3, false, false);
    }

    // D layout: acc[g] holds row (strip*16 + g + 8*half), col (n0 + m)
    float* o = support + (size_t)(strip * 16 + 8 * half) * GCN_OUT_F + m;
    #pragma unroll
    for (int g = 0; g < 8; ++g) {
        o[(size_t)g * GCN_OUT_F + 0 ] = acc0[g];
        o[(size_t)g * GCN_OUT_F + 16] = acc1[g];
        o[(size_t)g * GCN_OUT_F + 32] = acc2[g];
        o[(size_t)g * GCN_OUT_F + 48] = acc3[g];
    }
}

// ---------------------------------------------------------------------------
// Kernel 2: out[n][f] = bias[f]  (initialize accumulation target)
// ---------------------------------------------------------------------------
__global__ __launch_bounds__(256) void gcn_init_out(
    float* __restrict__ out, const float* __restrict__ bias)
{
    const int i = blockIdx.x * blockDim.x + threadIdx.x;
    if (i < GCN_N_NODES * GCN_OUT_F) {
        out[i] = bias[i & (GCN_OUT_F - 1)];
    }
}

// ---------------------------------------------------------------------------
// Kernel 3: COO scatter. One wave per edge; lane l handles features 2l,2l+1.
// 256 B fully-coalesced gather from support (L2-resident: 12.8 MB << 192 MB),
// then 2 hardware fp32 atomics into out[dst] (L2 atomic units).
// ---------------------------------------------------------------------------
__global__ __launch_bounds__(256) void gcn_scatter(
    const int*   __restrict__ edge_src,
    const int*   __restrict__ edge_dst,
    const float* __restrict__ edge_weight,
    const float* __restrict__ support,
    float*       __restrict__ out)
{
    const int wave = (blockIdx.x * blockDim.x + threadIdx.x) >> 5;
    const int lane = threadIdx.x & 31;
    if (wave >= GCN_N_EDGES) return;

    const int   s = edge_src[wave];
    const int   d = edge_dst[wave];
    const float w = edge_weight[wave];

    const v2f* sp = (const v2f*)(support + (size_t)s * GCN_OUT_F);
    v2f v = sp[lane];                                   // global_load_b64, coalesced

    float* op = out + (size_t)d * GCN_OUT_F + 2 * lane;
    __hip_atomic_fetch_add(op + 0, w * v.x, __ATOMIC_RELAXED, __HIP_MEMORY_SCOPE_AGENT);
    __hip_atomic_fetch_add(op + 1, w * v.y, __ATOMIC_RELAXED, __HIP_MEMORY_SCOPE_AGENT);
}

// ---------------------------------------------------------------------------
extern "C" void kernel_launch(void* const* d_in, const int* in_sizes, int n_in,
                              void* d_out, int out_size, void* d_ws, size_t ws_size,
                              hipStream_t stream)
{
    const float* x    = (const float*)d_in[0];   // [50000,128]
    const int*   esrc = (const int*)  d_in[1];   // [800000]
    const int*   edst = (const int*)  d_in[2];   // [800000]
    const float* ew   = (const float*)d_in[3];   // [800000]
    const float* W    = (const float*)d_in[4];   // [128,64]
    const float* bias = (const float*)d_in[5];   // [64]
    float* out     = (float*)d_out;              // [50000,64]
    float* support = (float*)d_ws;               // [50000,64] scratch (12.8 MB)

    // GEMM: 3125 strips, 8 waves (strips) per 256-thread block
    gcn_gemm_wmma<<<(GCN_M_TILES + 7) / 8, 256, 0, stream>>>(x, W, support);

    // out = bias (broadcast)
    gcn_init_out<<<(GCN_N_NODES * GCN_OUT_F + 255) / 256, 256, 0, stream>>>(out, bias);

    // scatter: one wave per edge
    gcn_scatter<<<(GCN_N_EDGES + 7) / 8, 256, 0, stream>>>(esrc, edst, ew, support, out);
}